// GraphAutoencoder_11312943857937
// MI455X (gfx1250) — compile-verified
//
#include <hip/hip_runtime.h>
#include <hip/hip_bf16.h>

typedef __attribute__((ext_vector_type(2))) float v2f;
typedef __attribute__((ext_vector_type(8))) float v8f;

#define NN   100000
#define EE   1600000
#define IND  128
#define HID  64
#define OUTD 128

// ---------------------------------------------------------------- atomics
__device__ __forceinline__ void atomAddF(float* p, float v) {
    unsafeAtomicAdd(p, v);   // lowers to global_atomic_add_f32 on gfx1250
}

// ---------------------------------------------------------------- degree / norm
__global__ void k_init_deg(float* deg, int n) {
    int i = blockIdx.x * blockDim.x + threadIdx.x;
    if (i < n) deg[i] = 1.0f;                       // self-loop weight
}

__global__ void k_deg_accum(const int* __restrict__ col, const float* __restrict__ w,
                            float* deg, int e) {
    int i = blockIdx.x * blockDim.x + threadIdx.x;
    if (i < e) atomAddF(&deg[col[i]], w[i]);
}

__global__ void k_finalize_dinv(float* dinv, int n) {
    int i = blockIdx.x * blockDim.x + threadIdx.x;
    if (i < n) dinv[i] = rsqrtf(dinv[i]);           // deg >= 1 always
}

__global__ void k_norm(const int* __restrict__ row, const int* __restrict__ col,
                       const float* __restrict__ w, const float* __restrict__ dinv,
                       float* __restrict__ norm, int e) {
    int i = blockIdx.x * blockDim.x + threadIdx.x;
    if (i < e) norm[i] = dinv[row[i]] * w[i] * dinv[col[i]];
}

// ---------------------------------------------------------------- WMMA GEMM
// C[M x NW] = A[M x K] * B[K x NW] (+ bias). One 16x16 tile per wave.
// f32 WMMA 16x16x4: A lanes 0-15 hold {K0,K1}, lanes 16-31 hold {K2,K3};
// B VGPR0/1 mirror the K split across lane halves; C/D: VGPR g -> M = g + 8*(lane>=16).
template <int K, int NW>
__global__ void k_gemm_wmma(const float* __restrict__ A, const float* __restrict__ B,
                            const float* __restrict__ bias, float* __restrict__ C) {
    const int lane  = threadIdx.x & 31;
    const int wave  = threadIdx.x >> 5;
    const int mbase = blockIdx.x * 16;
    const int nbase = wave * 16;
    const int hl    = lane >> 4;          // lane half: 0 or 1
    const int l15   = lane & 15;
    const int khalf = hl * 2;

    const float* arow = A + (size_t)(mbase + l15) * K;
    v8f acc = {0.f, 0.f, 0.f, 0.f, 0.f, 0.f, 0.f, 0.f};

#if __has_builtin(__builtin_amdgcn_wmma_f32_16x16x4_f32)
    for (int k = 0; k < K; k += 4) {
        v2f a, b;
        a.x = arow[k + khalf];
        a.y = arow[k + khalf + 1];
        b.x = B[(size_t)(k + khalf)     * NW + nbase + l15];
        b.y = B[(size_t)(k + khalf + 1) * NW + nbase + l15];
        acc = __builtin_amdgcn_wmma_f32_16x16x4_f32(
                  false, a, false, b, (short)0, acc, false, false);
    }
#else
    // scalar fallback (should not be taken on gfx1250)
    for (int k = 0; k < K; ++k) {
        float av = arow[k];
        #pragma unroll
        for (int g = 0; g < 8; ++g)
            acc[g] = fmaf(av, B[(size_t)k * NW + nbase + l15], acc[g]);
    }
#endif

    const int mrow = mbase + hl * 8;
    const int col  = nbase + l15;
    const float bb = bias ? bias[col] : 0.0f;
    #pragma unroll
    for (int g = 0; g < 8; ++g)
        C[(size_t)(mrow + g) * NW + col] = acc[g] + bb;
}

// ---------------------------------------------------------------- self-loop init / relu
// dst[i,:] = src[i,:] * dinv[i]^2   (self-loop message, norm = dinv^2)
__global__ void k_selfinit(const float* __restrict__ src, const float* __restrict__ dinv,
                           float* __restrict__ dst, int total) {
    int g = blockIdx.x * blockDim.x + threadIdx.x;
    if (g >= total) return;
    int i = g >> 6;                       // HID == 64
    float di = dinv[i];
    dst[g] = src[g] * di * di;
}

// a = relu(agg1 + b1); agg1 <- a;  agg2 <- a * dinv^2 (self-loop init for layer 2)
__global__ void k_relu_fuse(float* __restrict__ agg1, const float* __restrict__ b1,
                            const float* __restrict__ dinv, float* __restrict__ agg2,
                            int total) {
    int g = blockIdx.x * blockDim.x + threadIdx.x;
    if (g >= total) return;
    int i = g >> 6;
    int j = g & 63;
    float v = fmaxf(agg1[g] + b1[j], 0.0f);
    agg1[g] = v;
    float di = dinv[i];
    agg2[g] = v * di * di;
}

// ---------------------------------------------------------------- edge scatter
// One wave (32 lanes) per edge; each lane moves 2 floats (HID=64).
// dst[col[e],:] += src[row[e],:] * norm[e]
__global__ void k_scatter(const int* __restrict__ row, const int* __restrict__ col,
                          const float* __restrict__ norm,
                          const float* __restrict__ src, float* __restrict__ dst, int e) {
    int g    = blockIdx.x * blockDim.x + threadIdx.x;
    int eidx = g >> 5;
    int lane = g & 31;
    if (eidx >= e) return;
    int   r  = row[eidx];
    int   c  = col[eidx];
    float nm = norm[eidx];
    const float2 v = ((const float2*)(src + (size_t)r * HID))[lane];
    float* d = dst + (size_t)c * HID + lane * 2;
    atomAddF(d + 0, v.x * nm);
    atomAddF(d + 1, v.y * nm);
}

// ---------------------------------------------------------------- launch
extern "C" void kernel_launch(void* const* d_in, const int* in_sizes, int n_in,
                              void* d_out, int out_size, void* d_ws, size_t ws_size,
                              hipStream_t stream) {
    const float* x   = (const float*)d_in[0];   // [N,128]
    const int*   ei  = (const int*)  d_in[1];   // [2,E] : row = ei[0:E], col = ei[E:2E]
    const float* w   = (const float*)d_in[2];   // [E]
    const float* W1  = (const float*)d_in[3];   // [128,64]
    const float* b1  = (const float*)d_in[4];   // [64]
    const float* W2  = (const float*)d_in[5];   // [64,128]
    const float* b2  = (const float*)d_in[6];   // [128]
    float*       out = (float*)d_out;           // [N,128]

    const int n = in_sizes[0] / IND;            // 100000
    const int e = in_sizes[1] / 2;              // 1600000
    const int* row = ei;
    const int* col = ei + e;

    // workspace layout (floats)
    float* ws   = (float*)d_ws;
    float* dinv = ws;                                   // [N]
    float* norm = ws + n;                               // [E]
    float* h1   = ws + n + e;                           // [N,64] (later reused as agg2)
    float* agg1 = h1 + (size_t)n * HID;                 // [N,64]
    float* agg2 = h1;                                   // reuse

    const int T = 256;
    const int nh = n * HID;

    // 1) degree -> dinv
    k_init_deg<<<(n + T - 1) / T, T, 0, stream>>>(dinv, n);
    k_deg_accum<<<(e + T - 1) / T, T, 0, stream>>>(col, w, dinv, e);
    k_finalize_dinv<<<(n + T - 1) / T, T, 0, stream>>>(dinv, n);
    // 2) per-edge norm
    k_norm<<<(e + T - 1) / T, T, 0, stream>>>(row, col, w, dinv, norm, e);
    // 3) h1 = x @ W1   (N/16 = 6250 tiles; 4 waves/block cover NW=64)
    k_gemm_wmma<IND, HID><<<n / 16, (HID / 16) * 32, 0, stream>>>(x, W1, nullptr, h1);
    // 4) agg1 = self-loop init, then scatter-add edges
    k_selfinit<<<(nh + T - 1) / T, T, 0, stream>>>(h1, dinv, agg1, nh);
    k_scatter<<<((e * 32) + T - 1) / T, T, 0, stream>>>(row, col, norm, h1, agg1, e);
    // 5) a1 = relu(agg1 + b1); agg2 = a1 * dinv^2 (self-loop init for layer 2)
    k_relu_fuse<<<(nh + T - 1) / T, T, 0, stream>>>(agg1, b1, dinv, agg2, nh);
    // 6) aggregate layer-2 messages in 64-dim space (linearity of segment_sum)
    k_scatter<<<((e * 32) + T - 1) / T, T, 0, stream>>>(row, col, norm, agg1, agg2, e);
    // 7) out = agg2 @ W2 + b2   (8 waves/block cover NW=128)
    k_gemm_wmma<HID, OUTD><<<n / 16, (OUTD / 16) * 32, 0, stream>>>(agg2, W2, b2, out);
}